// FreqSparAtten_18562848653551
// MI455X (gfx1250) — compile-verified
//
#include <hip/hip_runtime.h>

// FreqSparAtten on MI455X (gfx1250, wave32).
// Memory-bound problem (~250MB traffic vs 4.3 GFLOP) -> use f32 WMMA
// (V_WMMA_F32_16X16X4_F32) to keep reference f32 numerics; matrix units are
// never the bottleneck here.

#define B_ 8
#define L_ 2048
#define H_ 16
#define E_ 128
#define S_ 256
#define BH_ (B_ * H_)
#define SCALE_ 0.08838834764831845f /* 1/sqrt(128) */

typedef float v2f __attribute__((ext_vector_type(2)));
typedef float v8f __attribute__((ext_vector_type(8)));

#define AS_ 132 /* padded LDS stride for Xq strip rows (avoid 64-bank conflicts) */
#define PS_ 260 /* padded LDS stride for score rows */

// ---------------------------------------------------------------------------
// Zero-fill out_ft region ([B,H,E,L] f32) with float4 stores.
// ---------------------------------------------------------------------------
__global__ void zero_kernel(float4* __restrict__ out, long n4) {
  long i = (long)blockIdx.x * blockDim.x + threadIdx.x;
  float4 z = {0.f, 0.f, 0.f, 0.f};
  if (i < n4) out[i] = z;
}

// ---------------------------------------------------------------------------
// Gather selected frequency rows: Xq[b,h,s,:] = q[b, iq[s], h, :]
//                                 Xk[b,h,s,:] = k[ikv[s], ...].
// One block per (b,h,s); 128 threads copy one contiguous 512B row each of q,k.
// ---------------------------------------------------------------------------
__global__ void gather_kernel(const float* __restrict__ q,
                              const float* __restrict__ k,
                              const int* __restrict__ iq,
                              const int* __restrict__ ikv,
                              float* __restrict__ Xq,
                              float* __restrict__ Xk) {
  int blk = blockIdx.x;
  int s = blk % S_;
  int h = (blk / S_) % H_;
  int b = blk / (S_ * H_);
  int e = threadIdx.x; // 0..127
  long src_q = (((long)b * L_ + iq[s]) * H_ + h) * E_ + e;
  long src_k = (((long)b * L_ + ikv[s]) * H_ + h) * E_ + e;
  long dst   = (((long)(b * H_ + h)) * S_ + s) * E_ + e;
  Xq[dst] = q[src_q];
  Xk[dst] = k[src_k];
}

// ---------------------------------------------------------------------------
// P = softmax(scale * Xq . Xk^T) per head, one block per (head, 16-row strip).
// 512 threads = 16 waves; wave w computes the 16x16 tile at columns w*16..+15
// with V_WMMA_F32_16X16X4_F32 over K=E=128 (32 WMMAs). Raw scores -> LDS,
// wave-per-row softmax (shfl_xor, wave32), result written to d_out xqk region.
// ---------------------------------------------------------------------------
__global__ void qk_softmax_kernel(const float* __restrict__ Xq,
                                  const float* __restrict__ Xk,
                                  float* __restrict__ P) {
  __shared__ float ldsA[16 * AS_];
  __shared__ float ldsS[16 * PS_];

  int strip = blockIdx.x & 15;        // S_/16 == 16 strips
  int bh    = blockIdx.x >> 4;
  int x0    = strip * 16;
  const float* Xqh = Xq + (long)bh * S_ * E_;
  const float* Xkh = Xk + (long)bh * S_ * E_;
  float*       Ph  = P  + (long)bh * S_ * S_;

  int tid = threadIdx.x; // 0..511
  // Stage the 16xE A-strip in LDS (shared by all 16 waves).
  for (int i = tid; i < 16 * E_; i += 512) {
    int r = i >> 7, c = i & (E_ - 1);
    ldsA[r * AS_ + c] = Xqh[(long)(x0 + r) * E_ + c];
  }
  __syncthreads();

  int wave = tid >> 5;            // tile column index 0..15
  int lane = tid & 31;
  int m    = lane & 15;           // A: M row / B: N col within tile
  int koff = (lane >> 4) << 1;    // K sub-offset 0 or 2 (ISA f32 A/B layout)
  const float* Bp = Xkh + (long)(wave * 16 + m) * E_;

  v8f acc = {0.f, 0.f, 0.f, 0.f, 0.f, 0.f, 0.f, 0.f};
  for (int kk = 0; kk < E_; kk += 4) {
    v2f a, bf;
    a.x  = ldsA[m * AS_ + kk + koff];
    a.y  = ldsA[m * AS_ + kk + koff + 1];
    bf.x = Bp[kk + koff];
    bf.y = Bp[kk + koff + 1];
    acc = __builtin_amdgcn_wmma_f32_16x16x4_f32(
        /*neg_a=*/false, a, /*neg_b=*/false, bf,
        /*c_mod=*/(short)0, acc, /*reuse_a=*/false, /*reuse_b=*/false);
  }

  // C layout: VGPR j -> row (lane<16 ? j : j+8), col = wave*16 + (lane&15).
  int mrow = (lane >> 4) << 3;
#pragma unroll
  for (int j = 0; j < 8; ++j)
    ldsS[(mrow + j) * PS_ + wave * 16 + m] = acc[j] * SCALE_;
  __syncthreads();

  // Softmax: wave r owns strip row r; 32 lanes x 8 cols each = 256.
  {
    int row = wave;
    float vals[8];
    float mx = -3.0e38f;
#pragma unroll
    for (int t = 0; t < 8; ++t) {
      vals[t] = ldsS[row * PS_ + lane + t * 32];
      mx = fmaxf(mx, vals[t]);
    }
#pragma unroll
    for (int off = 16; off >= 1; off >>= 1)
      mx = fmaxf(mx, __shfl_xor(mx, off, 32));
    float sum = 0.f;
#pragma unroll
    for (int t = 0; t < 8; ++t) {
      vals[t] = __expf(vals[t] - mx);
      sum += vals[t];
    }
#pragma unroll
    for (int off = 16; off >= 1; off >>= 1)
      sum += __shfl_xor(sum, off, 32);
    float inv = 1.f / sum;
#pragma unroll
    for (int t = 0; t < 8; ++t)
      Ph[(long)(x0 + row) * S_ + lane + t * 32] = vals[t] * inv;
  }
}

// ---------------------------------------------------------------------------
// D'[e,x] = sum_y Xk[y,e] * P[x,y] per head (M=e, N=x so lanes vary along x),
// then scatter column iq[x] of out_ft[b,h,e,:]. One block per (head, 16-e
// strip); 16 waves each own one 16(x)-tile; K=S=256 -> 64 WMMAs per tile.
// P is re-read from the xqk output region (L2-resident, 192MB L2).
// ---------------------------------------------------------------------------
__global__ void pv_scatter_kernel(const float* __restrict__ P,
                                  const float* __restrict__ Xk,
                                  const int* __restrict__ iq,
                                  float* __restrict__ out) {
  int estrip = blockIdx.x & 7;         // E_/16 == 8 strips
  int bh     = blockIdx.x >> 3;
  int e0     = estrip * 16;
  const float* Ph  = P  + (long)bh * S_ * S_;
  const float* Xkh = Xk + (long)bh * S_ * E_;
  float*       oh  = out + (long)bh * E_ * L_;

  int tid  = threadIdx.x; // 0..511
  int wave = tid >> 5;    // x-tile 0..15
  int lane = tid & 31;
  int n    = lane & 15;
  int koff = (lane >> 4) << 1;
  int x    = wave * 16 + n;      // N index (and A's M index column source)
  int em   = e0 + n;             // A fragment row (e)
  const float* Prow = Ph + (long)x * S_;

  v8f acc = {0.f, 0.f, 0.f, 0.f, 0.f, 0.f, 0.f, 0.f};
  for (int kk = 0; kk < S_; kk += 4) {
    v2f a, bf;
    a.x  = Xkh[(long)(kk + koff) * E_ + em];     // A[m=e][k=y] = Xk[y][e]
    a.y  = Xkh[(long)(kk + koff + 1) * E_ + em];
    bf.x = Prow[kk + koff];                      // B[k=y][n=x] = P[x][y]
    bf.y = Prow[kk + koff + 1];
    acc = __builtin_amdgcn_wmma_f32_16x16x4_f32(
        false, a, false, bf, (short)0, acc, false, false);
  }

  // C: VGPR j -> e = e0 + (lane<16 ? j : j+8), x = wave*16 + (lane&15).
  int iqx   = iq[x];
  int ebase = e0 + ((lane >> 4) << 3);
#pragma unroll
  for (int j = 0; j < 8; ++j)
    oh[(long)(ebase + j) * L_ + iqx] = acc[j];
}

// ---------------------------------------------------------------------------
extern "C" void kernel_launch(void* const* d_in, const int* in_sizes, int n_in,
                              void* d_out, int out_size, void* d_ws, size_t ws_size,
                              hipStream_t stream) {
  const float* q  = (const float*)d_in[0];
  const float* k  = (const float*)d_in[1];
  // d_in[2] = v (unused by reference), d_in[3] = attn_mask (unused)
  const int* iq  = (const int*)d_in[4];
  const int* ikv = (const int*)d_in[5];

  float* out = (float*)d_out;
  const long OUTFT = (long)BH_ * E_ * L_;     // out_ft elements
  float* P = out + OUTFT;                      // xqk output doubles as P buffer

  float* Xq = (float*)d_ws;                    // [BH, S, E]
  float* Xk = Xq + (long)BH_ * S_ * E_;        // [BH, S, E]  (~33.5MB total ws)

  long n4 = OUTFT / 4;
  zero_kernel<<<(int)((n4 + 255) / 256), 256, 0, stream>>>((float4*)out, n4);
  gather_kernel<<<BH_ * S_, 128, 0, stream>>>(q, k, iq, ikv, Xq, Xk);
  qk_softmax_kernel<<<BH_ * (S_ / 16), 512, 0, stream>>>(Xq, Xk, P);
  pv_scatter_kernel<<<BH_ * (E_ / 16), 512, 0, stream>>>(P, Xk, iq, out);
}